// Net_68023692034553
// MI455X (gfx1250) — compile-verified
//
#include <hip/hip_runtime.h>
#include <math.h>

typedef __attribute__((ext_vector_type(2))) float v2f;
typedef __attribute__((ext_vector_type(8))) float v8f;

#define NFEAT 500
#define HID   256
#define NCLS  40
#define KHOPS 10
#define LDS_STRIDE 260   // 256 + 4 pad: kills bank conflicts (row stride % 64 != 0)

// ---------------------------------------------------------------------------
// Fused MLP: h = relu(x@W1+b1) @ W2 + b2, one 16-row strip per workgroup.
// Phase 1: 4 waves x 4 n-tiles of GEMM1 via v_wmma_f32_16x16x4_f32 -> LDS.
// Phase 2: waves 0..2 do GEMM2 (48 cols, store first 40) from LDS.
// ---------------------------------------------------------------------------
__global__ void __launch_bounds__(128)
mlp_kernel(const float* __restrict__ x,  const float* __restrict__ W1,
           const float* __restrict__ b1, const float* __restrict__ W2,
           const float* __restrict__ b2, float* __restrict__ hout, int N)
{
    __shared__ float h1s[16 * LDS_STRIDE];

    const int lane  = threadIdx.x & 31;
    const int w     = threadIdx.x >> 5;          // wave id 0..3
    const int m     = lane & 15;                 // row-in-tile / col-in-tile
    const int koff  = (lane >> 4) * 2;           // K split across lane halves
    const int mrow8 = (lane >> 4) * 8;           // C/D: upper half-wave = M+8
    const int rowBase = blockIdx.x * 16;

    // ---- phase 1: GEMM1 (16 rows x 256 cols), K = 500 ----
    int row  = rowBase + m;
    int rowc = row < N ? row : N - 1;            // clamp reads for tail blocks
    const float* xr = x + (long)rowc * NFEAT;

    v8f acc[4];
    #pragma unroll
    for (int j = 0; j < 4; ++j) acc[j] = 0;

    for (int k0 = 0; k0 < NFEAT; k0 += 4) {
        v2f a;                                   // A 16x4 f32 fragment
        a.x = xr[k0 + koff];
        a.y = xr[k0 + koff + 1];
        const float* w1p = W1 + (long)(k0 + koff) * HID;
        #pragma unroll
        for (int j = 0; j < 4; ++j) {
            int col = w * 64 + j * 16 + m;       // B 4x16 f32 fragment
            v2f b;
            b.x = w1p[col];
            b.y = w1p[HID + col];
            acc[j] = __builtin_amdgcn_wmma_f32_16x16x4_f32(
                false, a, false, b, (short)0, acc[j], false, false);
        }
    }

    // bias + relu, stage hidden tile in LDS
    #pragma unroll
    for (int j = 0; j < 4; ++j) {
        int colg   = w * 64 + j * 16 + m;
        float bias = b1[colg];
        #pragma unroll
        for (int i = 0; i < 8; ++i) {
            float v = acc[j][i] + bias;
            v = fmaxf(v, 0.0f);
            h1s[(i + mrow8) * LDS_STRIDE + colg] = v;
        }
    }
    __syncthreads();

    // ---- phase 2: GEMM2 (16 rows x 48 cols, K = 256), waves 0..2 ----
    if (w < 3) {
        v8f c2 = 0;
        const int col = w * 16 + m;
        for (int k0 = 0; k0 < HID; k0 += 4) {
            v2f a;
            a.x = h1s[m * LDS_STRIDE + k0 + koff];
            a.y = h1s[m * LDS_STRIDE + k0 + koff + 1];
            v2f b = 0;
            if (col < NCLS) {
                b.x = W2[(k0 + koff) * NCLS + col];
                b.y = W2[(k0 + koff + 1) * NCLS + col];
            }
            c2 = __builtin_amdgcn_wmma_f32_16x16x4_f32(
                false, a, false, b, (short)0, c2, false, false);
        }
        if (col < NCLS) {
            float bias = b2[col];
            #pragma unroll
            for (int i = 0; i < 8; ++i) {
                int r = rowBase + i + mrow8;
                if (r < N) hout[(long)r * NCLS + col] = c2[i] + bias;
            }
        }
    }
}

// ---------------------------------------------------------------------------
// GCN norm helpers
// ---------------------------------------------------------------------------
__global__ void deg_init(float* __restrict__ deg, int N) {
    int i = blockIdx.x * blockDim.x + threadIdx.x;
    if (i < N) deg[i] = 1.0f;                    // self-loop contribution
}

__global__ void deg_count(const int* __restrict__ dst, float* __restrict__ deg, int E) {
    int e = blockIdx.x * blockDim.x + threadIdx.x;
    if (e < E) atomicAdd(&deg[dst[e]], 1.0f);
}

__global__ void deg_finish(float* __restrict__ dinv, int N) {
    int i = blockIdx.x * blockDim.x + threadIdx.x;
    if (i < N) {
        float d = dinv[i];
        dinv[i] = (d > 0.0f) ? rsqrtf(d) : 0.0f;
    }
}

// ---------------------------------------------------------------------------
// Propagation: h_next = D^-1/2 (A+I) D^-1/2 h  (self-loop init + edge scatter)
// ---------------------------------------------------------------------------
__global__ void spmm_self(const float* __restrict__ hin, float* __restrict__ hnext,
                          const float* __restrict__ dinv, int total) {
    int i = blockIdx.x * blockDim.x + threadIdx.x;
    if (i < total) {
        float di = dinv[i / NCLS];
        hnext[i] = di * di * hin[i];
    }
}

__global__ void spmm_edges(const int* __restrict__ src, const int* __restrict__ dst,
                           const float* __restrict__ dinv,
                           const float* __restrict__ hin, float* __restrict__ hnext, int E) {
    int e = blockIdx.x * blockDim.x + threadIdx.x;
    if (e >= E) return;
    int s = src[e], d = dst[e];
    float nrm = dinv[s] * dinv[d];
    const float4* hi = (const float4*)(hin + (long)s * NCLS);
    float* ho = hnext + (long)d * NCLS;
    #pragma unroll
    for (int g = 0; g < NCLS / 4; ++g) {
        float4 v = hi[g];
        atomicAdd(ho + 4 * g + 0, nrm * v.x);
        atomicAdd(ho + 4 * g + 1, nrm * v.y);
        atomicAdd(ho + 4 * g + 2, nrm * v.z);
        atomicAdd(ho + 4 * g + 3, nrm * v.w);
    }
}

// ---------------------------------------------------------------------------
// Streaming depth-attention: acc += sigmoid(h . proj_w + proj_b) * h
// ---------------------------------------------------------------------------
__global__ void combine(const float* __restrict__ h, float* __restrict__ acc,
                        const float* __restrict__ pw, const float* __restrict__ pb,
                        int N, int init) {
    int n = blockIdx.x * blockDim.x + threadIdx.x;
    if (n >= N) return;
    const float* hr = h + (long)n * NCLS;
    float t = pb[0];
    #pragma unroll
    for (int c = 0; c < NCLS; ++c) t += hr[c] * pw[c];
    float s = 1.0f / (1.0f + expf(-t));
    float* ar = acc + (long)n * NCLS;
    if (init) {
        #pragma unroll
        for (int c = 0; c < NCLS; ++c) ar[c] = s * hr[c];
    } else {
        #pragma unroll
        for (int c = 0; c < NCLS; ++c) ar[c] += s * hr[c];
    }
}

__global__ void log_softmax_rows(const float* __restrict__ acc, float* __restrict__ out, int N) {
    int n = blockIdx.x * blockDim.x + threadIdx.x;
    if (n >= N) return;
    const float* ar = acc + (long)n * NCLS;
    float mx = -3.402823e38f;
    #pragma unroll
    for (int c = 0; c < NCLS; ++c) mx = fmaxf(mx, ar[c]);
    float sum = 0.0f;
    #pragma unroll
    for (int c = 0; c < NCLS; ++c) sum += expf(ar[c] - mx);
    float lse = mx + logf(sum);
    float* orow = out + (long)n * NCLS;
    #pragma unroll
    for (int c = 0; c < NCLS; ++c) orow[c] = ar[c] - lse;
}

// ---------------------------------------------------------------------------
extern "C" void kernel_launch(void* const* d_in, const int* in_sizes, int n_in,
                              void* d_out, int out_size, void* d_ws, size_t ws_size,
                              hipStream_t stream) {
    const float* x  = (const float*)d_in[0];
    const int*   ei = (const int*)  d_in[1];
    const float* W1 = (const float*)d_in[2];
    const float* b1 = (const float*)d_in[3];
    const float* W2 = (const float*)d_in[4];
    const float* b2 = (const float*)d_in[5];
    const float* pw = (const float*)d_in[6];
    const float* pb = (const float*)d_in[7];

    const int N = in_sizes[0] / NFEAT;
    const int E = in_sizes[1] / 2;
    const int* src = ei;
    const int* dst = ei + E;

    // workspace layout (floats): dinv[N] | hA[N*40] | hB[N*40] | acc[N*40]
    float* ws   = (float*)d_ws;
    float* dinv = ws;
    float* hA   = ws + N;
    float* hB   = hA + (size_t)N * NCLS;
    float* accb = hB + (size_t)N * NCLS;
    float* out  = (float*)d_out;

    const int T = 256;
    dim3 blk(T);

    // 1) fused MLP -> hA
    mlp_kernel<<<(N + 15) / 16, 128, 0, stream>>>(x, W1, b1, W2, b2, hA, N);

    // 2) gcn_norm degree -> dinv (in place)
    deg_init  <<<(N + T - 1) / T, blk, 0, stream>>>(dinv, N);
    deg_count <<<(E + T - 1) / T, blk, 0, stream>>>(dst, dinv, E);
    deg_finish<<<(N + T - 1) / T, blk, 0, stream>>>(dinv, N);

    // 3) streaming attention combine, hop 0
    combine<<<(N + T - 1) / T, blk, 0, stream>>>(hA, accb, pw, pb, N, 1);

    // 4) K hops of normalized propagation + combine
    float* cur = hA;
    float* nxt = hB;
    const int totNC = N * NCLS;
    for (int k = 0; k < KHOPS; ++k) {
        spmm_self <<<(totNC + T - 1) / T, blk, 0, stream>>>(cur, nxt, dinv, totNC);
        spmm_edges<<<(E + T - 1) / T,     blk, 0, stream>>>(src, dst, dinv, cur, nxt, E);
        combine   <<<(N + T - 1) / T,     blk, 0, stream>>>(nxt, accb, pw, pb, N, 0);
        float* t = cur; cur = nxt; nxt = t;
    }

    // 5) log-softmax -> d_out
    log_softmax_rows<<<(N + T - 1) / T, blk, 0, stream>>>(accb, out, N);
}